// DWTLayer_38740605010209
// MI455X (gfx1250) — compile-verified
//
#include <hip/hip_runtime.h>

// CDNA5 / gfx1250 wave32 WMMA types
typedef __attribute__((ext_vector_type(2))) float v2f;   // A / B operand of V_WMMA_F32_16X16X4_F32
typedef __attribute__((ext_vector_type(8))) float v8f;   // C / D accumulator (16x16 f32)

constexpr int       C    = 256;
constexpr int       PIX  = 128 * 128;                    // pixels per (b, c) plane
constexpr long long CH   = PIX;                          // floats between channels
constexpr long long BSTR = (long long)C * PIX;           // floats between batches

// Select one of 4 filter taps (or 0) without a memory-indexed table.
__device__ __forceinline__ float tap_sel(int idx, float a, float b, float c, float d) {
    float v = 0.0f;
    v = (idx == 0) ? a : v;
    v = (idx == 1) ? b : v;
    v = (idx == 2) ? c : v;
    v = (idx == 3) ? d : v;
    return v;
}

// Process one 128-pixel strip for one 16-coefficient k-tile: 8 iterations of
// (9 B-chunk loads -> 18 WMMAs -> 16 NT stores). EDGE adds zero-pad predication.
template <bool EDGE>
__device__ __forceinline__ void dwt_strip(const float* __restrict__ loadbase,
                                          int c0_lane,
                                          float* __restrict__ stlo,
                                          float* __restrict__ sthi,
                                          const v2f* Alo, const v2f* Ahi) {
#pragma unroll 1
    for (int i = 0; i < 8; ++i) {                 // 8 x 16 pixels = one W row
        const int px = i * 16;
        v8f dlo = {}, dhi = {};
#pragma unroll
        for (int j = 0; j < 9; ++j) {
            // B-matrix (4x16): comp r at lane l holds B[r + 2*(l>>4), l&15].
            v2f b;
#pragma unroll
            for (int r = 0; r < 2; ++r) {
                const float* p = loadbase + (long long)(4 * j + r) * CH + px;
                if (EDGE) {
                    const int ch = c0_lane + 4 * j + r;   // zero-mode padding
                    b[r] = (ch >= 0 && ch < C) ? __builtin_nontemporal_load(p) : 0.0f;
                } else {
                    b[r] = __builtin_nontemporal_load(p);
                }
            }
            dlo = __builtin_amdgcn_wmma_f32_16x16x4_f32(false, Alo[j], false, b,
                                                        (short)0, dlo, false, false);
            dhi = __builtin_amdgcn_wmma_f32_16x16x4_f32(false, Ahi[j], false, b,
                                                        (short)0, dhi, false, false);
        }
        // D layout: comp r at lane l holds D[r + 8*(l>>4), l&15]; bases pre-offset.
#pragma unroll
        for (int r = 0; r < 8; ++r) {
            __builtin_nontemporal_store(dlo[r], stlo + (long long)r * CH + px);
            __builtin_nontemporal_store(dhi[r], sthi + (long long)r * CH + px);
        }
    }
}

// db2 channel-DWT as banded GEMM on the f32 WMMA pipe.
//   x, out: (16, 256, 128, 128) f32 ; out[:, :128] = lo, out[:, 128:] = hi
// One wave = one (batch, image-row, k-tile): 8 x (16 out-ch x 16 px) tiles.
__global__ __launch_bounds__(256) void dwt_db2_wmma_kernel(const float* __restrict__ x,
                                                           float* __restrict__ out) {
    const int lane = threadIdx.x & 31;
    const int wv   = threadIdx.x >> 5;                    // 8 waves / block (wave32)
    const int task = blockIdx.x * 8 + wv;                 // 16384 strip tasks
    const int ktile = task & 7;                           // 8 k-tiles of 16 coeffs
    const int strip = task >> 3;                          // 2048 strips (batch, h)
    const int batch = strip >> 7;                         // 128 rows per batch
    const int p0    = (strip & 127) << 7;                 // row start pixel

    const int m    = lane & 15;                           // A row / B,D column
    const int half = lane >> 4;                           // lane group 0-15 / 16-31

    // db2 analysis filters, pre-reversed as in the reference.
    const float h00 =  0.48296291314469025f, h01 =  0.8365163037378079f,
                h02 =  0.22414386804185735f, h03 = -0.12940952255092145f;
    const float h10 = -0.12940952255092145f, h11 = -0.22414386804185735f,
                h12 =  0.8365163037378079f,  h13 = -0.48296291314469025f;

    // Banded weight chunks: A[mrow, kwin] = h[kwin - 2*mrow] if 0 <= idx <= 3.
    // A-matrix VGPR layout (ISA 7.12.2): comp r at lane l holds A[l&15, r + 2*(l>>4)].
    v2f Alo[9], Ahi[9];
#pragma unroll
    for (int j = 0; j < 9; ++j) {
#pragma unroll
        for (int r = 0; r < 2; ++r) {
            const int kwin = 4 * j + (half * 2 + r);
            const int idx  = kwin - 2 * m;
            Alo[j][r] = tap_sel(idx, h00, h01, h02, h03);
            Ahi[j][r] = tap_sel(idx, h10, h11, h12, h13);
        }
    }
    // Pin the 18 constant matrices in VGPRs: forbid per-WMMA rematerialization
    // of the v_cndmask chains (seen in round-1 asm).
#pragma unroll
    for (int j = 0; j < 9; ++j) {
        asm volatile("" : "+v"(Alo[j]), "+v"(Ahi[j]));
    }

    const int K0 = ktile << 4;                  // first lo coefficient of tile
    const int c0 = 2 * K0 - 2;                  // first channel of 34-wide window

    const float* xb = x + (long long)batch * BSTR + p0 + m;       // m == column
    const float* loadbase = xb + (long long)(c0 + 2 * half) * CH; // + (4j+r)*CH
    float* ob   = out + (long long)batch * BSTR + p0 + m;
    float* stlo = ob + (long long)(K0 + 8 * half) * CH;           // + r*CH
    float* sthi = ob + (long long)(128 + K0 + 8 * half) * CH;     // + r*CH

    if (ktile == 0 || ktile == 7) {             // wave-uniform branch
        dwt_strip<true>(loadbase, c0 + 2 * half, stlo, sthi, Alo, Ahi);
    } else {
        dwt_strip<false>(loadbase, c0 + 2 * half, stlo, sthi, Alo, Ahi);
    }
}

extern "C" void kernel_launch(void* const* d_in, const int* in_sizes, int n_in,
                              void* d_out, int out_size, void* d_ws, size_t ws_size,
                              hipStream_t stream) {
    (void)in_sizes; (void)n_in; (void)d_ws; (void)ws_size; (void)out_size;
    const float* x = (const float*)d_in[0];
    float* out = (float*)d_out;
    // 16 batches * 128 rows * 8 k-tiles = 16384 wave tasks;
    // 8 waves (256 threads) per block -> 2048 blocks. Exact coverage, no tail.
    dwt_db2_wmma_kernel<<<dim3(2048), dim3(256), 0, stream>>>(x, out);
}